// FeatureSimilarity_41790031790867
// MI455X (gfx1250) — compile-verified
//
#include <hip/hip_runtime.h>
#include <hip/hip_bf16.h>

// ---------------------------------------------------------------------------
// FeatureSimilarity (negative L2 distance matrix) for MI455X / gfx1250.
//
// out[i][j] = -sqrt(max(||f_i||^2 + ||f_j||^2 - 2 <f_i, f_j>, 0))
//
// N = 8192, D = 1024, f32.  Compute-bound on V_WMMA_F32_16X16X4_F32 issue
// rate (137 GFLOP total vs ~11.5us of HBM store time for the 268MB output at
// 23.3 TB/s; the 32MB feature matrix is L2-resident on a 192MB L2).
//
// v2: exploit symmetry -- compute only upper-triangle 128x128 tiles (2080 of
// 4096 blocks) and mirror each off-diagonal tile via per-lane b128 stores
// (lane (h,m) of a C/D fragment holds 8 consecutive M values = 8 consecutive
// columns of the mirrored tile).  Halves the WMMA work.
// ---------------------------------------------------------------------------

typedef __attribute__((ext_vector_type(2))) float v2f;
typedef __attribute__((ext_vector_type(4))) float v4f;
typedef __attribute__((ext_vector_type(8))) float v8f;

#define NROWS 8192
#define NDIM  1024
#define BM 128            // output tile rows per workgroup
#define BN 128            // output tile cols per workgroup
#define BK 32             // K chunk staged in LDS
#define LDS_STRIDE 36     // 32 + 4 pad floats: rows stay 16B-aligned for b128
                          // stores; WMMA fragment reads hit 32 distinct banks
                          // (36*m mod 64 enumerates {4j}, +2 gives {4j+2}).

// --------------------------- row-norm kernel -------------------------------
__global__ __launch_bounds__(256)
void rownorm_kernel(const float* __restrict__ f, float* __restrict__ sq) {
    __shared__ float red[256];
    const int row = blockIdx.x;
    const float* rp = f + (size_t)row * NDIM;
    float s = 0.0f;
    #pragma unroll
    for (int i = 0; i < NDIM / 256; ++i) {
        float v = rp[threadIdx.x + 256 * i];
        s += v * v;
    }
    red[threadIdx.x] = s;
    __syncthreads();
    #pragma unroll
    for (int off = 128; off > 0; off >>= 1) {
        if (threadIdx.x < off) red[threadIdx.x] += red[threadIdx.x + off];
        __syncthreads();
    }
    if (threadIdx.x == 0) sq[row] = red[0];
}

// ------------------------- fused GEMM + epilogue ---------------------------
// 256 threads = 8 waves. Wave w computes a 16x128 strip of the 128x128 tile:
// 8 accumulators of 16x16 f32 (v8f each, per CDNA5 C/D layout).
__global__ __launch_bounds__(256)
void l2sim_kernel(const float* __restrict__ f,
                  const float* __restrict__ sq,
                  float* __restrict__ out) {
    // Upper-triangle tiles only (J >= I); lower tiles come from the mirror
    // stores of their transposed partners.
    if (blockIdx.x < blockIdx.y) return;

    __shared__ float As[BM * LDS_STRIDE];
    __shared__ float Bs[BN * LDS_STRIDE];

    const int I    = blockIdx.y * BM;
    const int J    = blockIdx.x * BN;
    const int tid  = threadIdx.x;
    const int wave = tid >> 5;        // 0..7
    const int lane = tid & 31;
    const int h    = lane >> 4;       // half-wave select (K pair / M+8)
    const int m    = lane & 15;       // row-in-fragment / column-in-tile

    v8f acc[8] = {};

    for (int k0 = 0; k0 < NDIM; k0 += BK) {
        __syncthreads();   // previous iteration's LDS reads done
        // Cooperative fill: 128 rows x 32 floats per matrix = 1024 float4
        // per matrix, 4 float4 per thread per matrix. Fully coalesced.
        #pragma unroll
        for (int i = 0; i < 4; ++i) {
            int idx = tid + 256 * i;
            int row = idx >> 3;       // 0..127
            int c4  = idx & 7;        // 0..7  (float4 column within chunk)
            v4f a = *(const v4f*)(f + (size_t)(I + row) * NDIM + k0 + 4 * c4);
            v4f b = *(const v4f*)(f + (size_t)(J + row) * NDIM + k0 + 4 * c4);
            *(v4f*)(&As[row * LDS_STRIDE + 4 * c4]) = a;
            *(v4f*)(&Bs[row * LDS_STRIDE + 4 * c4]) = b;
        }
        __syncthreads();

        // 8 k-steps of 4, 8 column tiles each -> 64 WMMAs per wave per chunk.
        // A fragment (16x4 f32): lanes 0-15 hold K=kk,kk+1 of row m,
        // lanes 16-31 hold K=kk+2,kk+3.  B = F^T tile has the identical
        // per-lane load pattern with row base J + 16n.
        #pragma unroll
        for (int kk = 0; kk < BK; kk += 4) {
            v2f a = *(const v2f*)(&As[(16 * wave + m) * LDS_STRIDE + kk + 2 * h]);
            #pragma unroll
            for (int n = 0; n < 8; ++n) {
                v2f b = *(const v2f*)(&Bs[(16 * n + m) * LDS_STRIDE + kk + 2 * h]);
                acc[n] = __builtin_amdgcn_wmma_f32_16x16x4_f32(
                    /*neg_a=*/false, a, /*neg_b=*/false, b,
                    /*c_mod=*/(short)0, acc[n],
                    /*reuse_a=*/false, /*reuse_b=*/false);
            }
        }
    }

    // Epilogue: C/D layout -> VGPR r holds (M = r + 8h, N = m) of each tile.
    // Normal tile: coalesced per-VGPR row stores.
    // Mirror tile (J,I): per lane, elements r=0..7 are 8 consecutive columns
    // starting at I + 16*wave + 8h -> two 16B stores per fragment.
    const bool mirror = (blockIdx.x != blockIdx.y);
    #pragma unroll
    for (int n = 0; n < 8; ++n) {
        const int col = J + 16 * n + m;
        const float sqj = sq[col];
        v8f res;
        #pragma unroll
        for (int r = 0; r < 8; ++r) {
            const int rowi = I + 16 * wave + r + 8 * h;
            float d2 = sq[rowi] + sqj - 2.0f * acc[n][r];
            d2 = fmaxf(d2, 0.0f);
            res[r] = -__builtin_sqrtf(d2);
            out[(size_t)rowi * NROWS + col] = res[r];
        }
        if (mirror) {
            float* mp = out + (size_t)col * NROWS + (I + 16 * wave + 8 * h);
            v4f lo = __builtin_shufflevector(res, res, 0, 1, 2, 3);
            v4f hi = __builtin_shufflevector(res, res, 4, 5, 6, 7);
            *(v4f*)(mp)     = lo;
            *(v4f*)(mp + 4) = hi;
        }
    }
}

// ------------------------------ launcher -----------------------------------
extern "C" void kernel_launch(void* const* d_in, const int* in_sizes, int n_in,
                              void* d_out, int out_size, void* d_ws, size_t ws_size,
                              hipStream_t stream) {
    const float* f  = (const float*)d_in[0];
    float* out      = (float*)d_out;
    float* sq       = (float*)d_ws;     // 8192 floats = 32KB of workspace

    rownorm_kernel<<<NROWS, 256, 0, stream>>>(f, sq);

    dim3 grid(NROWS / BN, NROWS / BM);  // 64 x 64 tiles; lower tri early-exits
    l2sim_kernel<<<grid, 256, 0, stream>>>(f, sq, out);
}